// Harmonic_3238405341533
// MI455X (gfx1250) — compile-verified
//
#include <hip/hip_runtime.h>
#include <math.h>

#define BATCH  4
#define FRAMES 400
#define DDIM   512
#define HARM   128
#define UPS    192
#define TLEN   (FRAMES * UPS)   // 76800
#define SRATE  48000.0f

typedef float v2f __attribute__((ext_vector_type(2)));
typedef float v8f __attribute__((ext_vector_type(8)));

__device__ __forceinline__ float mod_sigmoid(float x) {
    // 2 * sigmoid(x)^ln(10) + 1e-7
    float s = 1.0f / (1.0f + __expf(-x));
    float p = __powf(s, 2.302585093f);
    return 2.0f * p + 1e-7f;
}

// ---------------------------------------------------------------------------
// Kernel 1: frame-rate alphas GEMM via fp32 WMMA.
//   C(1600x128) = hidden(1600x512) @ W(512x128), then mod_sigmoid.
// One wave per 16x16 tile; block = 8 waves covering all 8 N-tiles (H=128).
// A frag (16x4 f32): lane m=lane&15; lanes<16 hold K=kk+0,1; lanes>=16 K=kk+2,3.
// B frag (4x16 f32): lane n=lane&15; same K-half split.
// C/D (16x16 f32): lane n=lane&15; VGPR v -> row m0+v (+8 for lanes>=16).
// ---------------------------------------------------------------------------
__global__ __launch_bounds__(256)
void alphas_gemm_kernel(const float* __restrict__ hidden,
                        const float* __restrict__ W,     // (D,H) row-major
                        const float* __restrict__ bias,  // (H)
                        float* __restrict__ Af)          // (B*F, H)
{
    const int lane = threadIdx.x & 31;
    const int wave = threadIdx.x >> 5;
    const int m0 = blockIdx.x * 16;          // 100 blocks -> 1600 rows
    const int n0 = wave * 16;                // 8 waves -> 128 cols
    const int m  = lane & 15;
    const int kh = (lane >> 4) << 1;         // 0 or 2
    const int n  = n0 + (lane & 15);

    const float* arow = hidden + (size_t)(m0 + m) * DDIM;
    v8f c = {0.f, 0.f, 0.f, 0.f, 0.f, 0.f, 0.f, 0.f};

    for (int kk = 0; kk < DDIM; kk += 4) {
        v2f a, b;
        a.x = arow[kk + kh];
        a.y = arow[kk + kh + 1];
        b.x = W[(size_t)(kk + kh)     * HARM + n];
        b.y = W[(size_t)(kk + kh + 1) * HARM + n];
        c = __builtin_amdgcn_wmma_f32_16x16x4_f32(false, a, false, b,
                                                  (short)0, c, false, false);
    }

    const int rbase = m0 + ((lane >> 4) << 3);   // +8 for upper lane half
    const float bn = bias[n];
    #pragma unroll
    for (int v = 0; v < 8; ++v) {
        int row = rbase + v;
        Af[(size_t)row * HARM + n] = mod_sigmoid(c[v] + bn);
    }
}

// ---------------------------------------------------------------------------
// Kernel 2: amplitude GEMV (1600 rows x 512) -> mod_sigmoid. One wave per row.
// ---------------------------------------------------------------------------
__global__ __launch_bounds__(256)
void amp_kernel(const float* __restrict__ hidden,
                const float* __restrict__ Wamp,  // (D)
                const float* __restrict__ bamp,  // (1)
                float* __restrict__ ampf)        // (B*F)
{
    const int lane = threadIdx.x & 31;
    const int wave = threadIdx.x >> 5;
    const int row  = blockIdx.x * 8 + wave;      // grid 200 blocks
    const float* hr = hidden + (size_t)row * DDIM;
    float acc = 0.f;
    for (int k = lane; k < DDIM; k += 32) acc += hr[k] * Wamp[k];
    for (int off = 16; off > 0; off >>= 1) acc += __shfl_down(acc, off, 32);
    if (lane == 0) ampf[row] = mod_sigmoid(acc + bamp[0]);
}

// ---------------------------------------------------------------------------
// Upsampled f0 (normalized by SR) at sample t.
// ---------------------------------------------------------------------------
__device__ __forceinline__ float f0u_at(const float* __restrict__ f0b, int t) {
    float coord = fmaxf((t + 0.5f) * (1.0f / UPS) - 0.5f, 0.0f);
    int   i0 = (int)coord;               // coord >= 0 -> trunc == floor
    float w  = coord - (float)i0;
    int   i1 = min(i0 + 1, FRAMES - 1);
    return (f0b[i0] + w * (f0b[i1] - f0b[i0])) * (1.0f / SRATE);
}

// ---------------------------------------------------------------------------
// Kernel 3: inclusive cumsum of f0u along T, one block (1024 thr) per batch.
// Hillis-Steele scan in LDS over 75 chunks with a running carry.
// ---------------------------------------------------------------------------
__global__ __launch_bounds__(1024)
void scan_kernel(const float* __restrict__ f0, float* __restrict__ phi)
{
    __shared__ float buf[1024];
    const int b = blockIdx.x;
    const int tid = threadIdx.x;
    const float* f0b = f0 + b * FRAMES;
    float* phib = phi + (size_t)b * TLEN;
    float carry = 0.f;

    for (int c = 0; c < TLEN / 1024; ++c) {
        int t = c * 1024 + tid;
        buf[tid] = f0u_at(f0b, t);
        __syncthreads();
        for (int off = 1; off < 1024; off <<= 1) {
            float x = (tid >= off) ? buf[tid - off] : 0.f;
            __syncthreads();
            buf[tid] += x;
            __syncthreads();
        }
        phib[t] = buf[tid] + carry;
        float tot = buf[1023];
        __syncthreads();
        carry += tot;
    }
}

// ---------------------------------------------------------------------------
// Kernel 4: per-sample synthesis. Block = (batch, frame) window of 192 t's.
// Lanes walk t (coalesced h-major alphas stores); h loop is sequential with
// the 3 relevant frame-alpha rows cached in LDS (broadcast reads).
// ---------------------------------------------------------------------------
__global__ __launch_bounds__(192)
void synth_kernel(const float* __restrict__ Af,    // (B*F,H)
                  const float* __restrict__ ampf,  // (B*F)
                  const float* __restrict__ phi,   // (B,T)
                  const float* __restrict__ f0,    // (B,F)
                  float* __restrict__ out)
{
    __shared__ float rows[3 * HARM];
    const int b = blockIdx.x / FRAMES;
    const int f = blockIdx.x % FRAMES;
    const int tid = threadIdx.x;
    const int base = max(f - 1, 0);

    for (int i = tid; i < 3 * HARM; i += 192) {
        int rr = min(base + i / HARM, FRAMES - 1);
        rows[i] = Af[(size_t)(b * FRAMES + rr) * HARM + (i & (HARM - 1))];
    }
    __syncthreads();

    const int t = f * UPS + tid;
    float coord = fmaxf((t + 0.5f) * (1.0f / UPS) - 0.5f, 0.0f);
    int   i0 = (int)coord;
    float w  = coord - (float)i0;
    int   i1 = min(i0 + 1, FRAMES - 1);

    const float* f0b = f0 + b * FRAMES;
    float f0u = (f0b[i0] + w * (f0b[i1] - f0b[i0])) * (1.0f / SRATE);
    const float* ab = ampf + b * FRAMES;
    float amp = ab[i0] + w * (ab[i1] - ab[i0]);
    float ph  = phi[(size_t)b * TLEN + t];

    const float* R0 = rows + (i0 - base) * HARM;
    const float* R1 = rows + (i1 - base) * HARM;

    float suma = 0.f, ssum = 0.f;
    #pragma unroll 4
    for (int h = 0; h < HARM; ++h) {
        float a  = R0[h] + w * (R1[h] - R0[h]);
        suma += a;
        float hi = (float)(h + 1);
        float aa = (f0u * hi < 0.5f) ? 1.0f : 0.0f;
        float z  = ph * hi;
        z -= floorf(z);                     // sin(2pi x) = sin(2pi frac(x))
        ssum += a * aa * __sinf(6.28318530718f * z);
    }
    float rn = 1.0f / suma;

    out[(size_t)b * TLEN + t] = ssum * rn * amp;                          // y
    out[(size_t)BATCH * TLEN + (size_t)b * TLEN + t] = amp;               // amplitude

    float* oal = out + (size_t)2 * BATCH * TLEN + (size_t)b * HARM * TLEN + t;
    #pragma unroll 4
    for (int h = 0; h < HARM; ++h) {
        float a = R0[h] + w * (R1[h] - R0[h]);
        oal[(size_t)h * TLEN] = a * rn;                                   // alphas
    }
}

// ---------------------------------------------------------------------------
extern "C" void kernel_launch(void* const* d_in, const int* in_sizes, int n_in,
                              void* d_out, int out_size, void* d_ws, size_t ws_size,
                              hipStream_t stream)
{
    const float* hidden = (const float*)d_in[0];   // (B,F,D)
    const float* f0     = (const float*)d_in[1];   // (B,F)
    const float* W_al   = (const float*)d_in[2];   // (D,H)
    const float* b_al   = (const float*)d_in[3];   // (H)
    const float* W_amp  = (const float*)d_in[4];   // (D,1)
    const float* b_amp  = (const float*)d_in[5];   // (1)
    float* out = (float*)d_out;

    float* ws   = (float*)d_ws;
    float* Af   = ws;                                        // B*F*H   = 204800
    float* ampf = Af + (size_t)BATCH * FRAMES * HARM;        // B*F     = 1600
    float* phi  = ampf + BATCH * FRAMES;                     // B*T     = 307200

    alphas_gemm_kernel<<<(BATCH * FRAMES) / 16, 256, 0, stream>>>(hidden, W_al, b_al, Af);
    amp_kernel<<<(BATCH * FRAMES) / 8, 256, 0, stream>>>(hidden, W_amp, b_amp, ampf);
    scan_kernel<<<BATCH, 1024, 0, stream>>>(f0, phi);
    synth_kernel<<<BATCH * FRAMES, 192, 0, stream>>>(Af, ampf, phi, f0, out);
}